// PathDevelopmentNetwork_43920335569122
// MI455X (gfx1250) — compile-verified
//
#include <hip/hip_runtime.h>
#include <math.h>

#define T_LEN 4096
#define BS 128
#define NCOL (BS * 9)          // 1152 columns = bs * 9 channels
#define CHUNK 128
#define NCHUNK (T_LEN / CHUNK) // 32

typedef __attribute__((ext_vector_type(2))) float v2f;
typedef __attribute__((ext_vector_type(8))) float v8f;

// ---------------------------------------------------------------- FNN tables
__device__ __forceinline__ float fnn_eval(float t, const float* W1, const float* b1,
                                          const float* W2, const float* b2) {
  float acc = b2[0];
#pragma unroll
  for (int h = 0; h < 5; ++h) acc += W2[h] * tanhf(W1[h] * t + b1[h]);
  return acc;
}

__global__ void fnn_tables_kernel(const float* fW1, const float* fb1, const float* fW2, const float* fb2,
                                  const float* pW1, const float* pb1, const float* pW2, const float* pb2,
                                  const float* gW1, const float* gb1, const float* gW2, const float* gb2,
                                  const float* qW1, const float* qb1, const float* qW2, const float* qb2,
                                  float4* __restrict__ Ctab) {
  int j = blockIdx.x * blockDim.x + threadIdx.x;
  if (j >= T_LEN) return;
  float tr = (float)(T_LEN - 1 - j);  // f_arr is time-reversed: coeff(d) = f(T-1-d)
  float td = (float)j;
  float4 v;
  v.x = fnn_eval(tr, fW1, fb1, fW2, fb2);   // Cf
  v.y = fnn_eval(tr, pW1, pb1, pW2, pb2);   // Cfp
  v.z = fnn_eval(td, gW1, gb1, gW2, gb2);   // Cg
  v.w = fnn_eval(td, qW1, qb1, qW2, qb2);   // Cgp
  Ctab[j] = v;
}

// ------------------------------------------------------- 3x3 SO(3) utilities
__device__ __forceinline__ void expm_skew(float a, float b, float c, float R[9]) {
  // M = [[0,a,b],[-a,0,c],[-b,-c,0]],  R = I + s*M + k*M^2  (Rodrigues)
  float th2 = a * a + b * b + c * c;
  float th = sqrtf(th2);
  float s, k;
  if (th > 1e-4f) {
    s = __sinf(th) / th;
    k = (1.0f - __cosf(th)) / th2;
  } else {
    s = 1.0f - th2 * (1.0f / 6.0f);
    k = 0.5f - th2 * (1.0f / 24.0f);
  }
  float aa = a * a, bb = b * b, cc = c * c, ab = a * b, ac = a * c, bc = b * c;
  R[0] = 1.0f - k * (aa + bb); R[1] =  s * a - k * bc;       R[2] =  s * b + k * ac;
  R[3] = -s * a - k * bc;      R[4] = 1.0f - k * (aa + cc);  R[5] =  s * c - k * ab;
  R[6] = -s * b + k * ac;      R[7] = -s * c - k * ab;       R[8] = 1.0f - k * (bb + cc);
}

__device__ __forceinline__ void mat3mul(const float A[9], const float B[9], float C[9]) {
#pragma unroll
  for (int r = 0; r < 3; ++r)
#pragma unroll
    for (int c = 0; c < 3; ++c)
      C[r * 3 + c] = A[r * 3 + 0] * B[0 * 3 + c] + A[r * 3 + 1] * B[1 * 3 + c] + A[r * 3 + 2] * B[2 * 3 + c];
}

__device__ __forceinline__ void skew_coeffs(const float* A, float* a, float* b, float* c) {
  // As = A - A^T (row-major 3x3): a=M01, b=M02, c=M12
  *a = A[1] - A[3]; *b = A[2] - A[6]; *c = A[5] - A[7];
}

// ------------------------------------------------ phase 1: per-chunk products
__global__ void chunk_prod_kernel(const float* __restrict__ x, const float* A1, const float* A2,
                                  float* __restrict__ P) {
  int idx = blockIdx.x * blockDim.x + threadIdx.x;
  if (idx >= BS * NCHUNK) return;
  int b = idx / NCHUNK, c = idx % NCHUNK;
  float as1, bs1, cs1, as2, bs2, cs2;
  skew_coeffs(A1, &as1, &bs1, &cs1);
  skew_coeffs(A2, &as2, &bs2, &cs2);
  float Pm[9] = {1, 0, 0, 0, 1, 0, 0, 0, 1};
  int t0 = c * CHUNK, t1 = t0 + CHUNK;
  for (int t = (t0 == 0 ? 1 : t0); t < t1; ++t) {  // scan uses E[1:]
    float z0 = x[((size_t)b * T_LEN + t) * 2 + 0];
    float z1 = x[((size_t)b * T_LEN + t) * 2 + 1];
    float E[9], Q[9];
    expm_skew(as1 * z0 + as2 * z1, bs1 * z0 + bs2 * z1, cs1 * z0 + cs2 * z1, E);
    mat3mul(Pm, E, Q);
#pragma unroll
    for (int i = 0; i < 9; ++i) Pm[i] = Q[i];
  }
#pragma unroll
  for (int i = 0; i < 9; ++i) P[(size_t)idx * 9 + i] = Pm[i];
}

// ---------------------------------------------- phase 2: exclusive chunk scan
__global__ void chunk_prefix_kernel(const float* __restrict__ P, float* __restrict__ S) {
  int b = blockIdx.x * blockDim.x + threadIdx.x;
  if (b >= BS) return;
  float Sm[9] = {1, 0, 0, 0, 1, 0, 0, 0, 1};
  for (int c = 0; c < NCHUNK; ++c) {
#pragma unroll
    for (int i = 0; i < 9; ++i) S[((size_t)b * NCHUNK + c) * 9 + i] = Sm[i];
    float Pc[9], Q[9];
#pragma unroll
    for (int i = 0; i < 9; ++i) Pc[i] = P[((size_t)b * NCHUNK + c) * 9 + i];
    mat3mul(Sm, Pc, Q);
#pragma unroll
    for (int i = 0; i < 9; ++i) Sm[i] = Q[i];
  }
}

// -------- phase 3: expand X; write X and X^T (== X^-1, SO3) in K-pair layout
// Layout: Xp[((k>>1)*NCOL + col)*2 + (k&1)]  -> lane B-fragment = one b64 load.
__global__ void write_X_kernel(const float* __restrict__ x, const float* A1, const float* A2,
                               const float* __restrict__ S, float* __restrict__ Xp,
                               float* __restrict__ Xip) {
  int idx = blockIdx.x * blockDim.x + threadIdx.x;
  if (idx >= BS * NCHUNK) return;
  int b = idx / NCHUNK, c = idx % NCHUNK;
  float as1, bs1, cs1, as2, bs2, cs2;
  skew_coeffs(A1, &as1, &bs1, &cs1);
  skew_coeffs(A2, &as2, &bs2, &cs2);
  float cur[9];
#pragma unroll
  for (int i = 0; i < 9; ++i) cur[i] = S[(size_t)idx * 9 + i];
  int t0 = c * CHUNK, t1 = t0 + CHUNK;
  for (int t = t0; t < t1; ++t) {
    if (t > 0) {
      float z0 = x[((size_t)b * T_LEN + t) * 2 + 0];
      float z1 = x[((size_t)b * T_LEN + t) * 2 + 1];
      float E[9], Q[9];
      expm_skew(as1 * z0 + as2 * z1, bs1 * z0 + bs2 * z1, cs1 * z0 + cs2 * z1, E);
      mat3mul(cur, E, Q);
#pragma unroll
      for (int i = 0; i < 9; ++i) cur[i] = Q[i];
    }
    size_t base = ((size_t)(t >> 1) * NCOL + (size_t)b * 9) * 2 + (t & 1);
#pragma unroll
    for (int r = 0; r < 3; ++r)
#pragma unroll
      for (int cc = 0; cc < 3; ++cc) {
        Xp[base + (r * 3 + cc) * 2]  = cur[r * 3 + cc];
        Xip[base + (r * 3 + cc) * 2] = cur[cc * 3 + r];  // inverse of rotation = transpose
      }
  }
}

// ------------------------- fused 4-way Toeplitz GEMM + elementwise combine
// out[s,n] = AA*AB + BA*BB,  AA = Tf @ Xinv, AB = Tg @ X, etc.
// Tf[s,k] = Cf[s-k] (k<=s).  One wave per 16-row s-tile x TWO 16-col n-tiles:
// A-fragment assembly (LDS + movs) amortized over 8 WMMAs; B traffic halved.
// Bulk loop (k0 < s0): all d>=1, no masking.  Diagonal block: clamped index +
// value-select (cndmask) so LDS loads stay unconditional and EXEC stays ~0.
__launch_bounds__(256)
__global__ void conv_gemm_kernel(const float4* __restrict__ Ctab,
                                 const float* __restrict__ Xp, const float* __restrict__ Xip,
                                 float* __restrict__ out) {
  __shared__ float4 ldsC[T_LEN];  // 64 KB: {Cf,Cfp,Cg,Cgp} interleaved
  int tid = threadIdx.x;
  for (int i = tid; i < T_LEN; i += 256) ldsC[i] = Ctab[i];
  __syncthreads();

  int wave = tid >> 5;
  int lane = tid & 31;
  int half = lane >> 4;   // WMMA fragment half (lanes 16-31)
  int lid  = lane & 15;
  int s0 = (blockIdx.x * 8 + wave) * 16;  // 16-row output tile owned by this wave
  int n0 = blockIdx.y * 32;               // two adjacent 16-col tiles
  int col = n0 + lid;
  int s = s0 + lid;       // A-fragment row (M = lane%16)

  v8f accAA0 = {}, accAB0 = {}, accBA0 = {}, accBB0 = {};
  v8f accAA1 = {}, accAB1 = {}, accBA1 = {}, accBB1 = {};

  // ---- bulk: k0 in [0, s0), step 4; trip count s0/4 is a multiple of 4 ----
  for (int k0 = 0; k0 < s0; k0 += 16) {
#pragma unroll
    for (int u = 0; u < 4; ++u) {
      int k  = k0 + 4 * u;
      int ka = k + 2 * half;           // even; fragment K indices ka, ka+1
      int d0 = s - ka;                 // >= 1 in bulk region
      float4 t0 = ldsC[d0];
      float4 t1 = ldsC[d0 - 1];
      v2f aF, aFp, aG, aGp;
      aF[0] = t0.x; aF[1] = t1.x;  aFp[0] = t0.y; aFp[1] = t1.y;
      aG[0] = t0.z; aG[1] = t1.z;  aGp[0] = t0.w; aGp[1] = t1.w;

      size_t boff = (size_t)(ka >> 1) * NCOL + col;
      const float2 bx0  = ((const float2*)Xp)[boff];
      const float2 bxi0 = ((const float2*)Xip)[boff];
      const float2 bx1  = ((const float2*)Xp)[boff + 16];
      const float2 bxi1 = ((const float2*)Xip)[boff + 16];
      v2f bX0, bXi0, bX1, bXi1;
      bX0[0] = bx0.x;  bX0[1] = bx0.y;   bXi0[0] = bxi0.x; bXi0[1] = bxi0.y;
      bX1[0] = bx1.x;  bX1[1] = bx1.y;   bXi1[0] = bxi1.x; bXi1[1] = bxi1.y;

      accAA0 = __builtin_amdgcn_wmma_f32_16x16x4_f32(false, aF,  false, bXi0, (short)0, accAA0, false, false);
      accBA0 = __builtin_amdgcn_wmma_f32_16x16x4_f32(false, aFp, false, bXi0, (short)0, accBA0, false, false);
      accAB0 = __builtin_amdgcn_wmma_f32_16x16x4_f32(false, aG,  false, bX0,  (short)0, accAB0, false, false);
      accBB0 = __builtin_amdgcn_wmma_f32_16x16x4_f32(false, aGp, false, bX0,  (short)0, accBB0, false, false);
      accAA1 = __builtin_amdgcn_wmma_f32_16x16x4_f32(false, aF,  false, bXi1, (short)0, accAA1, false, false);
      accBA1 = __builtin_amdgcn_wmma_f32_16x16x4_f32(false, aFp, false, bXi1, (short)0, accBA1, false, false);
      accAB1 = __builtin_amdgcn_wmma_f32_16x16x4_f32(false, aG,  false, bX1,  (short)0, accAB1, false, false);
      accBB1 = __builtin_amdgcn_wmma_f32_16x16x4_f32(false, aGp, false, bX1,  (short)0, accBB1, false, false);
    }
  }

  // ---- diagonal 16x16 block: k0 = s0 .. s0+12 (4 steps, masked by value) ----
#pragma unroll
  for (int u = 0; u < 4; ++u) {
    int k  = s0 + 4 * u;
    int ka = k + 2 * half;
    int d0 = s - ka;
    int d1 = d0 - 1;
    float4 t0 = ldsC[d0 > 0 ? d0 : 0];
    float4 t1 = ldsC[d1 > 0 ? d1 : 0];
    v2f aF, aFp, aG, aGp;
    aF[0]  = (d0 >= 0) ? t0.x : 0.0f;  aF[1]  = (d1 >= 0) ? t1.x : 0.0f;
    aFp[0] = (d0 >= 0) ? t0.y : 0.0f;  aFp[1] = (d1 >= 0) ? t1.y : 0.0f;
    aG[0]  = (d0 >= 0) ? t0.z : 0.0f;  aG[1]  = (d1 >= 0) ? t1.z : 0.0f;
    aGp[0] = (d0 >= 0) ? t0.w : 0.0f;  aGp[1] = (d1 >= 0) ? t1.w : 0.0f;

    size_t boff = (size_t)(ka >> 1) * NCOL + col;
    const float2 bx0  = ((const float2*)Xp)[boff];
    const float2 bxi0 = ((const float2*)Xip)[boff];
    const float2 bx1  = ((const float2*)Xp)[boff + 16];
    const float2 bxi1 = ((const float2*)Xip)[boff + 16];
    v2f bX0, bXi0, bX1, bXi1;
    bX0[0] = bx0.x;  bX0[1] = bx0.y;   bXi0[0] = bxi0.x; bXi0[1] = bxi0.y;
    bX1[0] = bx1.x;  bX1[1] = bx1.y;   bXi1[0] = bxi1.x; bXi1[1] = bxi1.y;

    accAA0 = __builtin_amdgcn_wmma_f32_16x16x4_f32(false, aF,  false, bXi0, (short)0, accAA0, false, false);
    accBA0 = __builtin_amdgcn_wmma_f32_16x16x4_f32(false, aFp, false, bXi0, (short)0, accBA0, false, false);
    accAB0 = __builtin_amdgcn_wmma_f32_16x16x4_f32(false, aG,  false, bX0,  (short)0, accAB0, false, false);
    accBB0 = __builtin_amdgcn_wmma_f32_16x16x4_f32(false, aGp, false, bX0,  (short)0, accBB0, false, false);
    accAA1 = __builtin_amdgcn_wmma_f32_16x16x4_f32(false, aF,  false, bXi1, (short)0, accAA1, false, false);
    accBA1 = __builtin_amdgcn_wmma_f32_16x16x4_f32(false, aFp, false, bXi1, (short)0, accBA1, false, false);
    accAB1 = __builtin_amdgcn_wmma_f32_16x16x4_f32(false, aG,  false, bX1,  (short)0, accAB1, false, false);
    accBB1 = __builtin_amdgcn_wmma_f32_16x16x4_f32(false, aGp, false, bX1,  (short)0, accBB1, false, false);
  }

  // ---- epilogue: out = AA*AB + BA*BB, layout (bs, T, 3, 3) ----
  int bcol0 = col / 9,        c90 = col % 9;
  int bcol1 = (col + 16) / 9, c91 = (col + 16) % 9;
#pragma unroll
  for (int v = 0; v < 8; ++v) {
    int srow = s0 + v + 8 * half;    // C/D layout: VGPR v -> M = v + 8*half
    out[((size_t)bcol0 * T_LEN + srow) * 9 + c90] = accAA0[v] * accAB0[v] + accBA0[v] * accBB0[v];
    out[((size_t)bcol1 * T_LEN + srow) * 9 + c91] = accAA1[v] * accAB1[v] + accBA1[v] * accBB1[v];
  }
}

// ---------------------------------------------------------------------------
extern "C" void kernel_launch(void* const* d_in, const int* in_sizes, int n_in,
                              void* d_out, int out_size, void* d_ws, size_t ws_size,
                              hipStream_t stream) {
  (void)in_sizes; (void)n_in; (void)out_size; (void)ws_size;
  const float* x  = (const float*)d_in[0];
  const float* A1 = (const float*)d_in[1];
  const float* A2 = (const float*)d_in[2];
  const float *fW1 = (const float*)d_in[3],  *fb1 = (const float*)d_in[4],
              *fW2 = (const float*)d_in[5],  *fb2 = (const float*)d_in[6];
  const float *pW1 = (const float*)d_in[7],  *pb1 = (const float*)d_in[8],
              *pW2 = (const float*)d_in[9],  *pb2 = (const float*)d_in[10];
  const float *gW1 = (const float*)d_in[11], *gb1 = (const float*)d_in[12],
              *gW2 = (const float*)d_in[13], *gb2 = (const float*)d_in[14];
  const float *qW1 = (const float*)d_in[15], *qb1 = (const float*)d_in[16],
              *qW2 = (const float*)d_in[17], *qb2 = (const float*)d_in[18];
  float* out = (float*)d_out;

  float* ws = (float*)d_ws;
  float4* Ctab = (float4*)ws;                          // 4096 float4
  float* P     = ws + 4 * T_LEN;                       // 128*32*9
  float* S     = P + (size_t)BS * NCHUNK * 9;          // 128*32*9
  float* Xp    = S + (size_t)BS * NCHUNK * 9;          // T*NCOL (K-pair layout)
  float* Xip   = Xp + (size_t)T_LEN * NCOL;            // T*NCOL (K-pair layout)

  fnn_tables_kernel<<<T_LEN / 256, 256, 0, stream>>>(
      fW1, fb1, fW2, fb2, pW1, pb1, pW2, pb2,
      gW1, gb1, gW2, gb2, qW1, qb1, qW2, qb2, Ctab);

  chunk_prod_kernel<<<(BS * NCHUNK) / 256, 256, 0, stream>>>(x, A1, A2, P);
  chunk_prefix_kernel<<<1, BS, 0, stream>>>(P, S);
  write_X_kernel<<<(BS * NCHUNK) / 256, 256, 0, stream>>>(x, A1, A2, S, Xp, Xip);

  dim3 grid(T_LEN / (8 * 16), NCOL / 32, 1);  // (32, 36)
  conv_gemm_kernel<<<grid, 256, 0, stream>>>(Ctab, Xp, Xip, out);
}